// Bert_sg_av_55138790146060
// MI455X (gfx1250) — compile-verified
//
#include <hip/hip_runtime.h>
#include <math.h>

#define B_  256
#define S_  512
#define V_  768
#define V4  192          // V_/4 (float4 columns)
#define H_  50
#define T_  513          // S+1 (mean row appended)
#define K2V 1536         // 2*V
#define NP  64           // H padded to multiple of 16 for WMMA

typedef __attribute__((ext_vector_type(2))) float v2f;
typedef __attribute__((ext_vector_type(8))) float v8f;

// workspace layout (float offsets; all float4-aligned)
#define OFF_M1  0
#define OFF_M2  (OFF_M1 + B_*V_)
#define OFF_C1  (OFF_M2 + B_*V_)
#define OFF_R   (OFF_C1 + B_*T_)
#define OFF_AX  (OFF_R  + B_*T_)
#define OFF_AY  (OFF_AX + B_*T_)
#define OFF_ATX (OFF_AY + B_*T_)
#define OFF_ATY (OFF_ATX + B_*V_)
#define OFF_O1  (OFF_ATY + B_*V_)
#define OFF_O2  (OFF_O1  + B_*NP)

// ---- pass 1: per-batch means over s (m1, m2); float4 streaming --------------
__global__ __launch_bounds__(192)
void k_means(const float* __restrict__ out1, const float* __restrict__ out2,
             float* __restrict__ ws) {
  int c = threadIdx.x;                  // float4 column 0..191
  int b = blockIdx.x;
  const float4* x = (const float4*)((blockIdx.y == 0) ? out1 : out2);
  float4* m = (float4*)(ws + ((blockIdx.y == 0) ? OFF_M1 : OFF_M2));
  const float4* p = x + (size_t)b * S_ * V4 + c;
  float ax = 0.f, ay = 0.f, az = 0.f, aw = 0.f;
  #pragma unroll 8
  for (int s = 0; s < S_; ++s) {
    float4 t = p[(size_t)s * V4];
    ax += t.x; ay += t.y; az += t.z; aw += t.w;
  }
  float4 r; r.x = ax * (1.0f/S_); r.y = ay * (1.0f/S_);
  r.z = az * (1.0f/S_); r.w = aw * (1.0f/S_);
  m[b * V4 + c] = r;
}

// ---- pass 2: score vectors  c1[b,s]=x1[b,s]·m2[b],  r[b,t]=m1[b]·x2[b,t] ----
__global__ __launch_bounds__(256)
void k_scores(const float* __restrict__ out1, const float* __restrict__ out2,
              float* __restrict__ ws) {
  int lane = threadIdx.x & 31;
  int wave = threadIdx.x >> 5;
  int s = blockIdx.x * 8 + wave;
  if (s >= T_) return;
  int b = blockIdx.y;
  const float *row, *vec; float* dst;
  if (blockIdx.z == 0) {
    row = (s < S_) ? out1 + (size_t)b * S_ * V_ + (size_t)s * V_
                   : ws + OFF_M1 + b * V_;
    vec = ws + OFF_M2 + b * V_;
    dst = ws + OFF_C1;
  } else {
    row = (s < S_) ? out2 + (size_t)b * S_ * V_ + (size_t)s * V_
                   : ws + OFF_M2 + b * V_;
    vec = ws + OFF_M1 + b * V_;
    dst = ws + OFF_R;
  }
  const float4* r4 = (const float4*)row;
  const float4* v4 = (const float4*)vec;
  float acc = 0.f;
  #pragma unroll
  for (int i = lane; i < V4; i += 32) {
    float4 a = r4[i], c = v4[i];
    acc += a.x * c.x + a.y * c.y + a.z * c.z + a.w * c.w;
  }
  #pragma unroll
  for (int off = 16; off > 0; off >>= 1) acc += __shfl_xor(acc, off);
  if (lane == 0) dst[b * T_ + s] = acc;
}

// ---- pass 3a: ax[b,:] = softmax_s(c1[b,:])  (513 values per batch) ----------
__global__ __launch_bounds__(256)
void k_softmax_s(float* __restrict__ ws) {
  __shared__ float red[256];
  int tid = threadIdx.x, b = blockIdx.x;
  const float* c1 = ws + OFF_C1 + b * T_;
  float* ax = ws + OFF_AX + b * T_;
  float v0 = c1[tid];
  float v1 = c1[tid + 256];
  float v2 = (tid == 0) ? c1[512] : -INFINITY;
  float mx = fmaxf(fmaxf(v0, v1), v2);
  red[tid] = mx; __syncthreads();
  for (int st = 128; st > 0; st >>= 1) {
    if (tid < st) red[tid] = fmaxf(red[tid], red[tid + st]);
    __syncthreads();
  }
  mx = red[0]; __syncthreads();
  float e0 = __expf(v0 - mx), e1 = __expf(v1 - mx);
  float e2 = (tid == 0) ? __expf(v2 - mx) : 0.f;
  red[tid] = e0 + e1 + e2; __syncthreads();
  for (int st = 128; st > 0; st >>= 1) {
    if (tid < st) red[tid] += red[tid + st];
    __syncthreads();
  }
  float inv = 1.0f / red[0];
  ax[tid] = e0 * inv; ax[tid + 256] = e1 * inv;
  if (tid == 0) ax[512] = e2 * inv;
}

// ---- pass 3b: ay[:,t] = softmax over batch axis of r[:,t] -------------------
__global__ __launch_bounds__(256)
void k_softmax_b(float* __restrict__ ws) {
  __shared__ float red[256];
  int b = threadIdx.x;   // B_ == 256
  int t = blockIdx.x;    // 0..512
  float v = ws[OFF_R + b * T_ + t];
  red[b] = v; __syncthreads();
  for (int st = 128; st > 0; st >>= 1) {
    if (b < st) red[b] = fmaxf(red[b], red[b + st]);
    __syncthreads();
  }
  float mx = red[0]; __syncthreads();
  float e = __expf(v - mx);
  red[b] = e; __syncthreads();
  for (int st = 128; st > 0; st >>= 1) {
    if (b < st) red[b] += red[b + st];
    __syncthreads();
  }
  ws[OFF_AY + b * T_ + t] = e / red[0];
}

// ---- pass 4: attx[b,v]=Σ_s ax·x1,  atty[b,v]=Σ_t ay·x2; float4 streaming ----
__global__ __launch_bounds__(192)
void k_weighted(const float* __restrict__ out1, const float* __restrict__ out2,
                float* __restrict__ ws) {
  __shared__ float w[T_];
  int tid = threadIdx.x;                // float4 column 0..191
  int b = blockIdx.x;
  const float *x, *m, *wsrc; float* dst;
  if (blockIdx.y == 0) { x = out1; m = ws + OFF_M1 + b * V_; wsrc = ws + OFF_AX + b * T_; dst = ws + OFF_ATX; }
  else                 { x = out2; m = ws + OFF_M2 + b * V_; wsrc = ws + OFF_AY + b * T_; dst = ws + OFF_ATY; }
  for (int i = tid; i < T_; i += 192) w[i] = wsrc[i];
  __syncthreads();
  const float4* p = (const float4*)x + (size_t)b * S_ * V4 + tid;
  float ax = 0.f, ay = 0.f, az = 0.f, aw = 0.f;
  #pragma unroll 4
  for (int s = 0; s < S_; ++s) {
    float ww = w[s];
    float4 t = p[(size_t)s * V4];
    ax += ww * t.x; ay += ww * t.y; az += ww * t.z; aw += ww * t.w;
  }
  float wm = w[512];
  float4 mv = ((const float4*)m)[tid];
  float4 r;
  r.x = ax + wm * mv.x; r.y = ay + wm * mv.y;
  r.z = az + wm * mv.z; r.w = aw + wm * mv.w;
  ((float4*)dst)[b * V4 + tid] = r;
}

// ---- pass 5: O1/O2 (256x50) = [mean || attn] (256x1536) @ Wg^T, via WMMA ----
// V_WMMA_F32_16X16X4_F32, one wave per 16-row M tile, 4 N tiles (50 -> 64).
// K loop split at the mean/attention boundary (uniform), N-pad guard hoisted.
__device__ __forceinline__ void wmma_half_k(
    const float* __restrict__ a1p, const float* __restrict__ a2p,
    const float* __restrict__ w0, const float* __restrict__ w1,
    const float* __restrict__ w2, const float* __restrict__ w3,
    float sel3, int koff, v8f acc1[4], v8f acc2[4]) {
  for (int k = 0; k < V_; k += 4) {
    int ka = k + koff;
    v2f a1 = *(const v2f*)(a1p + ka);
    v2f a2 = *(const v2f*)(a2p + ka);
    v2f b0 = *(const v2f*)(w0 + ka);
    v2f b1 = *(const v2f*)(w1 + ka);
    v2f b2 = *(const v2f*)(w2 + ka);
    v2f b3 = *(const v2f*)(w3 + ka);
    b3.x *= sel3; b3.y *= sel3;
    acc1[0] = __builtin_amdgcn_wmma_f32_16x16x4_f32(false, a1, false, b0, (short)0, acc1[0], false, false);
    acc2[0] = __builtin_amdgcn_wmma_f32_16x16x4_f32(false, a2, false, b0, (short)0, acc2[0], false, false);
    acc1[1] = __builtin_amdgcn_wmma_f32_16x16x4_f32(false, a1, false, b1, (short)0, acc1[1], false, false);
    acc2[1] = __builtin_amdgcn_wmma_f32_16x16x4_f32(false, a2, false, b1, (short)0, acc2[1], false, false);
    acc1[2] = __builtin_amdgcn_wmma_f32_16x16x4_f32(false, a1, false, b2, (short)0, acc1[2], false, false);
    acc2[2] = __builtin_amdgcn_wmma_f32_16x16x4_f32(false, a2, false, b2, (short)0, acc2[2], false, false);
    acc1[3] = __builtin_amdgcn_wmma_f32_16x16x4_f32(false, a1, false, b3, (short)0, acc1[3], false, false);
    acc2[3] = __builtin_amdgcn_wmma_f32_16x16x4_f32(false, a2, false, b3, (short)0, acc2[3], false, false);
  }
}

__global__ __launch_bounds__(32)
void k_out_gemm(const float* __restrict__ Wg, const float* __restrict__ bg,
                float* __restrict__ ws) {
  int lane = threadIdx.x;
  int mt   = blockIdx.x;              // 0..15
  int arow = mt * 16 + (lane & 15);   // A row this lane loads
  int koff = (lane >> 4) * 2;         // K sub-offset {0,2}
  int ncl  = lane & 15;               // N within tile

  v8f acc1[4], acc2[4];
  #pragma unroll
  for (int nt = 0; nt < 4; ++nt) {
    int n = nt * 16 + ncl;
    float bias = (n < H_) ? bg[n] : 0.f;   // N-dependent only: same for all 8 rows
    #pragma unroll
    for (int j = 0; j < 8; ++j) { acc1[nt][j] = bias; acc2[nt][j] = bias; }
  }

  // B-operand row pointers (N-tiles 0..2 always in range; tile 3 clamped+masked)
  const float* w0 = Wg + (size_t)(0 * 16 + ncl) * K2V;
  const float* w1 = Wg + (size_t)(1 * 16 + ncl) * K2V;
  const float* w2 = Wg + (size_t)(2 * 16 + ncl) * K2V;
  int n3 = 48 + ncl;
  const float* w3 = Wg + (size_t)((n3 < H_) ? n3 : (H_ - 1)) * K2V;
  float sel3 = (n3 < H_) ? 1.f : 0.f;

  // first half of K: mean rows (m1 / m2)
  wmma_half_k(ws + OFF_M1 + (size_t)arow * V_, ws + OFF_M2 + (size_t)arow * V_,
              w0, w1, w2, w3, sel3, koff, acc1, acc2);
  // second half of K: attention rows (atty / attx), Wg columns offset by V_
  wmma_half_k(ws + OFF_ATY + (size_t)arow * V_, ws + OFF_ATX + (size_t)arow * V_,
              w0 + V_, w1 + V_, w2 + V_, w3 + V_, sel3, koff, acc1, acc2);

  float* o1 = ws + OFF_O1;
  float* o2 = ws + OFF_O2;
  #pragma unroll
  for (int nt = 0; nt < 4; ++nt) {
    int col = nt * 16 + ncl;
    #pragma unroll
    for (int j = 0; j < 8; ++j) {
      int row = mt * 16 + j + 8 * (lane >> 4);
      o1[row * NP + col] = acc1[nt][j];
      o2[row * NP + col] = acc2[nt][j];
    }
  }
}

// ---- pass 6: tiny MLP head + sigmoid, one thread per batch ------------------
__global__ __launch_bounds__(256)
void k_head(const float* __restrict__ Wfd, const float* __restrict__ bfd,
            const float* __restrict__ Wff, const float* __restrict__ bff,
            const float* __restrict__ ws, float* __restrict__ outp) {
  int b = threadIdx.x;
  const float* o1 = ws + OFF_O1 + b * NP;
  const float* o2 = ws + OFF_O2 + b * NP;
  float logit = bff[0];
  for (int j = 0; j < H_; ++j) {
    float hj = bfd[j];
    const float* wr = Wfd + j * 2 * H_;
    for (int i = 0; i < H_; ++i) hj += wr[i] * o1[i] + wr[H_ + i] * o2[i];
    hj = fmaxf(hj, 0.f);
    logit += Wff[j] * hj;
  }
  outp[b] = 1.0f / (1.0f + __expf(-logit));
}

extern "C" void kernel_launch(void* const* d_in, const int* in_sizes, int n_in,
                              void* d_out, int out_size, void* d_ws, size_t ws_size,
                              hipStream_t stream) {
  const float* out1 = (const float*)d_in[0];
  const float* out2 = (const float*)d_in[1];
  const float* Wg   = (const float*)d_in[2];
  const float* bg   = (const float*)d_in[3];
  const float* Wfd  = (const float*)d_in[4];
  const float* bfd  = (const float*)d_in[5];
  const float* Wff  = (const float*)d_in[6];
  const float* bff  = (const float*)d_in[7];
  float* ws   = (float*)d_ws;
  float* outp = (float*)d_out;

  k_means    <<<dim3(B_, 2), 192, 0, stream>>>(out1, out2, ws);
  k_scores   <<<dim3((T_+7)/8, B_, 2), 256, 0, stream>>>(out1, out2, ws);
  k_softmax_s<<<B_, 256, 0, stream>>>(ws);
  k_softmax_b<<<T_, 256, 0, stream>>>(ws);
  k_weighted <<<dim3(B_, 2), 192, 0, stream>>>(out1, out2, ws);
  k_out_gemm <<<16, 32, 0, stream>>>(Wg, bg, ws);
  k_head     <<<1, 256, 0, stream>>>(Wfd, bfd, Wff, bff, ws, outp);
}